// NeRFRenderer_90640989815202
// MI455X (gfx1250) — compile-verified
//
#include <hip/hip_runtime.h>

typedef float v2f __attribute__((ext_vector_type(2)));
typedef float v8f __attribute__((ext_vector_type(8)));

namespace {

constexpr int KC = 64;
constexpr int KF = 64;
constexpr int WAVES = 8; // rays per block (1 wave per ray)

struct RayScratch {
  float zc[64];     // coarse z (sorted by construction)
  float zf[64];     // fine z (unsorted)
  float zcomb[128]; // merged sorted z
  float cdf[68];    // cdf[0..64]
  float r0[128];    // sigmoid rgb channel 0 per sample
  float r1[128];
  float r2[128];
  float sg[128];    // relu(sigma) per sample
};

__device__ __forceinline__ v8f wmma4(v2f a, v2f b, v8f c) {
  // D = A(16x4,f32) * B(4x16,f32) + C(16x16,f32)
  return __builtin_amdgcn_wmma_f32_16x16x4_f32(false, a, false, b, (short)0, c,
                                               false, false);
}

__device__ __forceinline__ float fast_rcp(float x) {
  return __builtin_amdgcn_rcpf(x); // v_rcp_f32 (~1 ulp)
}

__device__ __forceinline__ float sigmoidf(float x) {
  return fast_rcp(1.0f + __expf(-x));
}

// Single-instruction ReLU (avoids fmaxf's canonicalize+max pair).
__device__ __forceinline__ float relu1(float x) {
  float r;
  asm("v_max_num_f32 %0, %1, 0" : "=v"(r) : "v"(x));
  return r;
}

} // namespace

__global__ __launch_bounds__(256) void nerf_fused_kernel(
    const float* __restrict__ rays,
    const float* __restrict__ u_coarse,
    const float* __restrict__ u_fine,
    const float* __restrict__ u_jitter,
    const float* __restrict__ W1,  // (3,32) row-major
    const float* __restrict__ b1,  // (32)
    const float* __restrict__ W2,  // (32,4) row-major
    const float* __restrict__ b2,  // (4)
    float* __restrict__ out,       // (B,8)
    int B)
{
  __shared__ RayScratch smem[WAVES];

  const int lane = threadIdx.x & 31;
  const int wid  = threadIdx.x >> 5;
  const int ray  = blockIdx.x * WAVES + wid;
  if (ray >= B) return; // wave-uniform exit
  RayScratch& L = smem[wid];

  const int  m       = lane & 15;
  const bool lowHalf = lane < 16;

  // Hidden-unit permutation so layer-1 D register pairs == layer-2 B operands.
  const int lperm = (m < 8) ? (4 * (m >> 1) + (m & 1))
                            : (4 * ((m - 8) >> 1) + 2 + ((m - 8) & 1));

  // Layer-1 A operands: A = W1^T (rows = hidden, K = x,y,z,bias).
  v2f Alo, Ahi;
  {
    const int hl = lperm;
    const int hh = 16 + lperm;
    Alo.x = lowHalf ? W1[0 * 32 + hl] : W1[2 * 32 + hl];
    Alo.y = lowHalf ? W1[1 * 32 + hl] : b1[hl];
    Ahi.x = lowHalf ? W1[0 * 32 + hh] : W1[2 * 32 + hh];
    Ahi.y = lowHalf ? W1[1 * 32 + hh] : b1[hh];
  }
  // Layer-2 A operands: A = W2^T chunks (rows = out channel 0..3, pad to 16).
  v2f A2[8];
#pragma unroll
  for (int j = 0; j < 8; ++j) {
    float x = 0.f, y = 0.f;
    if (m < 4) {
      const int rbase = 4 * j + (lowHalf ? 0 : 2);
      x = W2[rbase * 4 + m];
      y = W2[(rbase + 1) * 4 + m];
    }
    A2[j].x = x;
    A2[j].y = y;
  }
  // Output bias (uniform scalars; added in the epilogue instead of via C).
  const float b2_0 = b2[0], b2_1 = b2[1], b2_2 = b2[2], b2_3 = b2[3];

  // Ray data (uniform across wave).
  const float* R = rays + ray * 8;
  const float ox = R[0], oy = R[1], oz = R[2];
  const float dx = R[3], dy = R[4], dz = R[5];
  const float nr = R[6], fr = R[7];

  // ---- Phase A: coarse stratified samples ----
#pragma unroll
  for (int t = 0; t < 2; ++t) {
    const int s = lane + 32 * t;
    const float u  = u_coarse[ray * KC + s];
    const float zs = ((float)s + u) * (1.0f / 64.0f);
    L.zc[s] = nr * (1.0f - zs) + fr * zs;
  }

  // ---- MLP over a 16-point group ----
  auto mlp_group = [&](const float* zarr, int g) {
    const float z = zarr[g * 16 + m];
    v2f Bp;
    if (lowHalf) { Bp.x = ox + z * dx; Bp.y = oy + z * dy; }
    else         { Bp.x = oz + z * dz; Bp.y = 1.0f; }
    v8f zero8;
#pragma unroll
    for (int i = 0; i < 8; ++i) zero8[i] = 0.f;
    v8f Dlo = wmma4(Alo, Bp, zero8);
    v8f Dhi = wmma4(Ahi, Bp, zero8);
#pragma unroll
    for (int i = 0; i < 8; ++i) {
      Dlo[i] = relu1(Dlo[i]);
      Dhi[i] = relu1(Dhi[i]);
    }
    v8f O = zero8;
#pragma unroll
    for (int j = 0; j < 4; ++j) {
      v2f Bc; Bc.x = Dlo[2 * j]; Bc.y = Dlo[2 * j + 1];
      O = wmma4(A2[j], Bc, O);
    }
#pragma unroll
    for (int j = 0; j < 4; ++j) {
      v2f Bc; Bc.x = Dhi[2 * j]; Bc.y = Dhi[2 * j + 1];
      O = wmma4(A2[4 + j], Bc, O);
    }
    if (lowHalf) {
      const int s = g * 16 + m;
      L.r0[s] = sigmoidf(O[0] + b2_0);
      L.r1[s] = sigmoidf(O[1] + b2_1);
      L.r2[s] = sigmoidf(O[2] + b2_2);
      L.sg[s] = relu1(O[3] + b2_3);
    }
  };

  // ---- Phase B: coarse MLP ----
#pragma unroll
  for (int g = 0; g < 4; ++g) mlp_group(L.zc, g);

  // ---- Phase C: coarse composite (lane owns samples 2l, 2l+1) ----
  const int s0 = 2 * lane, s1 = s0 + 1;
  const float zc0 = L.zc[s0], zc1 = L.zc[s1];
  float zc2 = fr;
  if (lane != 31) zc2 = L.zc[s1 + 1];
  const float a0 = 1.0f - __expf(-(zc1 - zc0) * L.sg[s0]);
  const float a1 = 1.0f - __expf(-(zc2 - zc1) * L.sg[s1]);
  const float f0 = 1.0f - a0 + 1e-10f;
  const float f1 = 1.0f - a1 + 1e-10f;
  float incl = f0 * f1;
#pragma unroll
  for (int i = 1; i < 32; i <<= 1) {
    const float u = __shfl_up(incl, i, 32);
    if (lane >= i) incl *= u;
  }
  float excl = __shfl_up(incl, 1, 32);
  if (lane == 0) excl = 1.0f;
  const float w0 = a0 * excl;
  const float w1 = a1 * excl * f0;
  float cr = w0 * L.r0[s0] + w1 * L.r0[s1];
  float cg = w0 * L.r1[s0] + w1 * L.r1[s1];
  float cb = w0 * L.r2[s0] + w1 * L.r2[s1];
  float cd = w0 * zc0 + w1 * zc1;
#pragma unroll
  for (int i = 16; i >= 1; i >>= 1) {
    cr += __shfl_xor(cr, i, 32);
    cg += __shfl_xor(cg, i, 32);
    cb += __shfl_xor(cb, i, 32);
    cd += __shfl_xor(cd, i, 32);
  }

  // ---- Phase D: fine sampling via inverse CDF ----
  const float wp0 = w0 + 1e-5f, wp1 = w1 + 1e-5f;
  float tot = wp0 + wp1;
#pragma unroll
  for (int i = 16; i >= 1; i >>= 1) tot += __shfl_xor(tot, i, 32);
  const float invTot = fast_rcp(tot);
  float sinc = wp0 + wp1;
#pragma unroll
  for (int i = 1; i < 32; i <<= 1) {
    const float u = __shfl_up(sinc, i, 32);
    if (lane >= i) sinc += u;
  }
  float sexc = __shfl_up(sinc, 1, 32);
  if (lane == 0) sexc = 0.0f;
  L.cdf[s0] = sexc * invTot;
  L.cdf[s1] = (sexc + wp0) * invTot;
  if (lane == 31) L.cdf[64] = sinc * invTot;

  float zf0 = 0.f, zf1 = 0.f;
#pragma unroll
  for (int t = 0; t < 2; ++t) {
    const int f = s0 + t;
    const float u = u_fine[ray * KF + f];
    int lo = 0, hi = 65; // count of cdf entries <= u  (searchsorted 'right')
    while (lo < hi) {
      const int mid = (lo + hi) >> 1;
      if (L.cdf[mid] <= u) lo = mid + 1; else hi = mid;
    }
    const float ind = fmaxf((float)lo - 1.0f, 0.0f);
    const float zs  = (ind + u_jitter[ray * KF + f]) * (1.0f / 64.0f);
    const float z   = nr * (1.0f - zs) + fr * zs;
    if (t == 0) zf0 = z; else zf1 = z;
    L.zf[f] = z;
  }

  // ---- Phase E: merge-rank sort of 64 coarse (sorted) + 64 fine ----
  int cLt0 = 0, cLt1 = 0, rk0 = 0, rk1 = 0;
  for (int j = 0; j < 64; ++j) {
    const float v = L.zf[j];
    cLt0 += (v < zc0) ? 1 : 0;
    cLt1 += (v < zc1) ? 1 : 0;
    rk0 += ((v < zf0) || (v == zf0 && j < s0)) ? 1 : 0;
    rk1 += ((v < zf1) || (v == zf1 && j < s1)) ? 1 : 0;
  }
  auto count_le_coarse = [&](float v) {
    int lo = 0, hi = 64;
    while (lo < hi) {
      const int mid = (lo + hi) >> 1;
      if (L.zc[mid] <= v) lo = mid + 1; else hi = mid;
    }
    return lo;
  };
  const int cc0 = count_le_coarse(zf0);
  const int cc1 = count_le_coarse(zf1);
  L.zcomb[s0 + cLt0] = zc0;
  L.zcomb[s1 + cLt1] = zc1;
  L.zcomb[rk0 + cc0] = zf0;
  L.zcomb[rk1 + cc1] = zf1;

  // ---- Phase F: fine MLP over 128 combined samples ----
#pragma unroll
  for (int g = 0; g < 8; ++g) mlp_group(L.zcomb, g);

  // ---- Phase G: fine composite (lane owns samples 4l..4l+3) ----
  const int b4 = 4 * lane;
  float z4[4], f4[4], a4[4];
#pragma unroll
  for (int i = 0; i < 4; ++i) z4[i] = L.zcomb[b4 + i];
  float zlast = fr;
  if (lane != 31) zlast = L.zcomb[b4 + 4];
  float locp = 1.0f;
#pragma unroll
  for (int i = 0; i < 4; ++i) {
    const float znext = (i < 3) ? z4[i + 1] : zlast;
    const float al = 1.0f - __expf(-(znext - z4[i]) * L.sg[b4 + i]);
    a4[i] = al;
    f4[i] = 1.0f - al + 1e-10f;
    locp *= f4[i];
  }
  float incl2 = locp;
#pragma unroll
  for (int i = 1; i < 32; i <<= 1) {
    const float u = __shfl_up(incl2, i, 32);
    if (lane >= i) incl2 *= u;
  }
  float T = __shfl_up(incl2, 1, 32);
  if (lane == 0) T = 1.0f;
  float fro = 0.f, fgo = 0.f, fbo = 0.f, fdo = 0.f;
#pragma unroll
  for (int i = 0; i < 4; ++i) {
    const float w = a4[i] * T;
    fro += w * L.r0[b4 + i];
    fgo += w * L.r1[b4 + i];
    fbo += w * L.r2[b4 + i];
    fdo += w * z4[i];
    T *= f4[i];
  }
#pragma unroll
  for (int i = 16; i >= 1; i >>= 1) {
    fro += __shfl_xor(fro, i, 32);
    fgo += __shfl_xor(fgo, i, 32);
    fbo += __shfl_xor(fbo, i, 32);
    fdo += __shfl_xor(fdo, i, 32);
  }

  if (lane == 0) {
    float* o = out + ray * 8;
    o[0] = fro; o[1] = fgo; o[2] = fbo; o[3] = fdo;
    o[4] = cr;  o[5] = cg;  o[6] = cb;  o[7] = cd;
  }
}

extern "C" void kernel_launch(void* const* d_in, const int* in_sizes, int n_in,
                              void* d_out, int out_size, void* d_ws, size_t ws_size,
                              hipStream_t stream) {
  (void)n_in; (void)out_size; (void)d_ws; (void)ws_size;
  const float* rays     = (const float*)d_in[0];
  const float* u_coarse = (const float*)d_in[1];
  const float* u_fine   = (const float*)d_in[2];
  const float* u_jitter = (const float*)d_in[3];
  const float* W1 = (const float*)d_in[4];
  const float* b1 = (const float*)d_in[5];
  const float* W2 = (const float*)d_in[6];
  const float* b2 = (const float*)d_in[7];
  float* out = (float*)d_out;

  const int B = in_sizes[0] / 8;
  const int grid = (B + WAVES - 1) / WAVES;
  nerf_fused_kernel<<<grid, 256, 0, stream>>>(rays, u_coarse, u_fine, u_jitter,
                                              W1, b1, W2, b2, out, B);
}